// GM_SAGE_13572096655879
// MI455X (gfx1250) — compile-verified
//
#include <hip/hip_runtime.h>

typedef __attribute__((ext_vector_type(2))) float v2f;
typedef __attribute__((ext_vector_type(8))) float v8f;

// ---------------------------------------------------------------------------
// Degree histogram: one thread per edge, native f32 atomic add into deg[dst].
// ---------------------------------------------------------------------------
__global__ void k_deg(const long long* __restrict__ dst, float* __restrict__ deg, int E) {
    int e = blockIdx.x * blockDim.x + threadIdx.x;
    if (e < E) unsafeAtomicAdd(&deg[dst[e]], 1.0f);
}

// ---------------------------------------------------------------------------
// Edge scatter-add: one wave (32 lanes) per edge; each lane moves float4
// (lane*16B -> full 512B row coalesced) and does 4 native f32 atomics.
// Both h[src] rows and agg[dst] rows are resident in the 192MB L2.
// ---------------------------------------------------------------------------
__global__ void k_scatter(const float* __restrict__ h,
                          const long long* __restrict__ src,
                          const long long* __restrict__ dst,
                          float* __restrict__ agg, int E) {
    int wave = (blockIdx.x * blockDim.x + threadIdx.x) >> 5;
    int lane = threadIdx.x & 31;
    if (wave >= E) return;
    long long s = src[wave];
    long long d = dst[wave];
    const float4 v = *(const float4*)(h + s * 128 + lane * 4);
    float* p = agg + d * 128 + lane * 4;
    unsafeAtomicAdd(p + 0, v.x);
    unsafeAtomicAdd(p + 1, v.y);
    unsafeAtomicAdd(p + 2, v.z);
    unsafeAtomicAdd(p + 3, v.w);
}

// ---------------------------------------------------------------------------
// Fused SAGE layer GEMM:
//   out = relu( (agg * 1/max(deg,1)) @ Wl + h @ Wr + bl + br )
// One wave computes a full 16x128 output stripe (8 x 16x16 tiles) so the A
// fragments (agg/h rows) are loaded ONCE per k-step and fed to 16 independent
// v_wmma_f32_16x16x4_f32 ops (8 tiles x 2 operand matrices) -> deep XDL ILP,
// 8x less A-side L2 traffic.
// A frag : lane holds A[lane%16][k + (lane/16)*2 + {0,1}]   (float2 load)
// B frag : lane holds W[k + (lane/16)*2 + {0,1}][tile*16 + lane%16]
// D frag : VGPR v holds D[v + 8*(lane/16)][tile*16 + lane%16]
// ---------------------------------------------------------------------------
__global__ void k_sage_gemm(const float* __restrict__ agg,
                            const float* __restrict__ deg,
                            const float* __restrict__ h,
                            const float* __restrict__ Wl, const float* __restrict__ bl,
                            const float* __restrict__ Wr, const float* __restrict__ br,
                            float* __restrict__ out, int nRowTiles, int N) {
    int wave = (blockIdx.x * blockDim.x + threadIdx.x) >> 5;
    int lane = threadIdx.x & 31;
    if (wave >= nRowTiles) return;

    int n    = lane & 15;                       // col within a 16-wide tile
    int kgrp = lane >> 4;                       // 0 or 1
    int row  = wave * 16 + n;                   // A row for this lane
    int rowL = row < N ? row : (N - 1);         // clamp loads (stores guarded)

    float invd = 1.0f / fmaxf(deg[rowL], 1.0f);

    const float* aggRow = agg + (size_t)rowL * 128 + kgrp * 2;
    const float* hRow   = h   + (size_t)rowL * 128 + kgrp * 2;

    v8f c[8] = {};
    for (int k = 0; k < 128; k += 4) {
        float2 ta = *(const float2*)(aggRow + k);
        float2 th = *(const float2*)(hRow + k);
        v2f aL; aL.x = ta.x * invd; aL.y = ta.y * invd;
        v2f aR; aR.x = th.x;        aR.y = th.y;
        int kk = k + kgrp * 2;
        const float* wl0 = Wl + kk * 128 + n;
        const float* wl1 = Wl + (kk + 1) * 128 + n;
        const float* wr0 = Wr + kk * 128 + n;
        const float* wr1 = Wr + (kk + 1) * 128 + n;
        #pragma unroll
        for (int t = 0; t < 8; ++t) {
            v2f bL; bL.x = wl0[t * 16]; bL.y = wl1[t * 16];
            v2f bR; bR.x = wr0[t * 16]; bR.y = wr1[t * 16];
            c[t] = __builtin_amdgcn_wmma_f32_16x16x4_f32(false, aL, false, bL, (short)0, c[t], false, false);
            c[t] = __builtin_amdgcn_wmma_f32_16x16x4_f32(false, aR, false, bR, (short)0, c[t], false, false);
        }
    }

    int dRowBase = wave * 16 + kgrp * 8;        // D rows this lane-half covers
    bool full = (wave * 16 + 15) < N;           // wave-uniform fast path
    #pragma unroll
    for (int t = 0; t < 8; ++t) {
        int col = t * 16 + n;
        float bias = bl[col] + br[col];
        if (full) {
            #pragma unroll
            for (int v = 0; v < 8; ++v) {
                float x = c[t][v] + bias;
                out[(size_t)(dRowBase + v) * 128 + col] = x > 0.0f ? x : 0.0f;
            }
        } else {
            #pragma unroll
            for (int v = 0; v < 8; ++v) {
                int r = dRowBase + v;
                if (r < N) {
                    float x = c[t][v] + bias;
                    out[(size_t)r * 128 + col] = x > 0.0f ? x : 0.0f;
                }
            }
        }
    }
}

// ---------------------------------------------------------------------------
// Final projection: out[N,40] = h2 @ W_out + b_out.  One wave = one row tile,
// all 3 column tiles (last partial: B loads clamped to col 39 so EXEC stays
// all-1s for the WMMAs; only stores masked).
// ---------------------------------------------------------------------------
__global__ void k_out_gemm(const float* __restrict__ h,
                           const float* __restrict__ W, const float* __restrict__ b,
                           float* __restrict__ out, int nRowTiles, int N) {
    int wave = (blockIdx.x * blockDim.x + threadIdx.x) >> 5;
    int lane = threadIdx.x & 31;
    if (wave >= nRowTiles) return;

    int n    = lane & 15;
    int kgrp = lane >> 4;
    int row  = wave * 16 + n;
    int rowL = row < N ? row : (N - 1);

    const float* hRow = h + (size_t)rowL * 128 + kgrp * 2;

    v8f c[3] = {};
    for (int k = 0; k < 128; k += 4) {
        float2 th = *(const float2*)(hRow + k);
        v2f a; a.x = th.x; a.y = th.y;
        int kk = k + kgrp * 2;
        #pragma unroll
        for (int t = 0; t < 3; ++t) {
            int col = t * 16 + n;
            int colL = col < 40 ? col : 39;     // clamp loads, mask stores
            v2f bb; bb.x = W[kk * 40 + colL]; bb.y = W[(kk + 1) * 40 + colL];
            c[t] = __builtin_amdgcn_wmma_f32_16x16x4_f32(false, a, false, bb, (short)0, c[t], false, false);
        }
    }

    int dRowBase = wave * 16 + kgrp * 8;
    bool fullRows = (wave * 16 + 15) < N;
    #pragma unroll
    for (int t = 0; t < 3; ++t) {
        int col = t * 16 + n;
        if (col >= 40) continue;                // lane-uniform within half-tiles
        float bias = b[col];
        #pragma unroll
        for (int v = 0; v < 8; ++v) {
            int r = dRowBase + v;
            if (fullRows || r < N) {
                out[(size_t)r * 40 + col] = c[t][v] + bias;
            }
        }
    }
}

// ---------------------------------------------------------------------------
extern "C" void kernel_launch(void* const* d_in, const int* in_sizes, int n_in,
                              void* d_out, int out_size, void* d_ws, size_t ws_size,
                              hipStream_t stream) {
    const float*     x     = (const float*)d_in[0];
    const long long* ei    = (const long long*)d_in[1];  // int64, shape (2, E)
    const float*     W_l0  = (const float*)d_in[2];
    const float*     b_l0  = (const float*)d_in[3];
    const float*     W_r0  = (const float*)d_in[4];
    const float*     b_r0  = (const float*)d_in[5];
    const float*     W_l1  = (const float*)d_in[6];
    const float*     b_l1  = (const float*)d_in[7];
    const float*     W_r1  = (const float*)d_in[8];
    const float*     b_r1  = (const float*)d_in[9];
    const float*     W_out = (const float*)d_in[10];
    const float*     b_out = (const float*)d_in[11];
    float* out = (float*)d_out;

    const int N = in_sizes[0] / 128;
    const int E = in_sizes[1] / 2;
    const long long* src = ei;
    const long long* dst = ei + E;

    // Carve workspace (256B aligned chunks).
    auto align256 = [](size_t v) { return (v + 255) & ~(size_t)255; };
    char* ws = (char*)d_ws;
    size_t off = 0;
    float* deg = (float*)(ws + off); off += align256((size_t)N * sizeof(float));
    float* agg = (float*)(ws + off); off += align256((size_t)N * 128 * sizeof(float));
    float* h1  = (float*)(ws + off); off += align256((size_t)N * 128 * sizeof(float));
    float* h2  = (float*)(ws + off); off += align256((size_t)N * 128 * sizeof(float));
    (void)ws_size; (void)n_in; (void)out_size;

    const int nRowTiles = (N + 15) / 16;

    // ---- degree (shared by both layers) ----
    hipMemsetAsync(deg, 0, (size_t)N * sizeof(float), stream);
    hipMemsetAsync(agg, 0, (size_t)N * 128 * sizeof(float), stream);
    {
        int threads = 256;
        int blocks = (E + threads - 1) / threads;
        k_deg<<<blocks, threads, 0, stream>>>(dst, deg, E);
    }

    // ---- layer 0: scatter x, fused GEMM -> h1 ----
    {
        int threads = 256;                       // 8 waves/block, 1 edge/wave
        int blocks = (E + 7) / 8;
        k_scatter<<<blocks, threads, 0, stream>>>(x, src, dst, agg, E);
    }
    {
        int threads = 256;                       // 8 row-tile waves per block
        int blocks = (nRowTiles + 7) / 8;
        k_sage_gemm<<<blocks, threads, 0, stream>>>(agg, deg, x, W_l0, b_l0, W_r0, b_r0,
                                                    h1, nRowTiles, N);
    }

    // ---- layer 1: scatter h1, fused GEMM -> h2 ----
    hipMemsetAsync(agg, 0, (size_t)N * 128 * sizeof(float), stream);
    {
        int threads = 256;
        int blocks = (E + 7) / 8;
        k_scatter<<<blocks, threads, 0, stream>>>(h1, src, dst, agg, E);
    }
    {
        int threads = 256;
        int blocks = (nRowTiles + 7) / 8;
        k_sage_gemm<<<blocks, threads, 0, stream>>>(agg, deg, h1, W_l1, b_l1, W_r1, b_r1,
                                                    h2, nRowTiles, N);
    }

    // ---- output projection ----
    {
        int threads = 256;
        int blocks = (nRowTiles + 7) / 8;
        k_out_gemm<<<blocks, threads, 0, stream>>>(h2, W_out, b_out, out, nRowTiles, N);
    }
}